// NoisyLinear_31825707664106
// MI455X (gfx1250) — compile-verified
//
#include <hip/hip_runtime.h>

// CDNA5 wave32 fp32 WMMA GEMM for NoisyLinear:
//   Out[M,N] = X[M,K] @ (W + Wn)^T[K,N] + (bias + bias_noise)
// M=2048, N=4096, K=4096, fp32 throughout (matches fp32 reference).
//
// Data movement:
//   A (x) tile      : GLOBAL_LOAD_ASYNC_TO_LDS_B128 (ASYNCcnt path, no VGPR staging)
//   B (W+Wn) tile   : global_load_b128 x2 -> v_pk_add_f32 -> ds_store (fused add)
//   W/Wn streams    : global_prefetch_b8 at distance 2 K-tiles
// Compute: v_wmma_f32_16x16x4_f32, 64 per wave per K-tile.

typedef float v2f __attribute__((ext_vector_type(2)));
typedef float v4f __attribute__((ext_vector_type(4)));
typedef float v8f __attribute__((ext_vector_type(8)));

#define M_DIM 2048
#define N_DIM 4096
#define K_DIM 4096

#define BM 128          // workgroup C-tile rows
#define BN 256          // workgroup C-tile cols
#define BK 16           // K tile depth
#define LDT (BK + 4)    // padded LDS row stride (floats): stride 20 -> conflict-free
#define NTHREADS 256    // 8 waves of 32

// Async copy of 16 bytes global -> LDS, per lane. Tracked by ASYNCcnt.
// LDS destination = low 32 bits of the flat shared-memory address
// (flat shared addr = {aperture_hi, wave-relative LDS offset}).
__device__ __forceinline__ void async_copy_b128(const float* gptr, float* lptr) {
    unsigned ldsoff = (unsigned)(size_t)lptr;
    unsigned long long ga = (unsigned long long)(size_t)gptr;
    asm volatile("global_load_async_to_lds_b128 %0, %1, off"
                 :: "v"(ldsoff), "v"(ga)
                 : "memory");
}

__device__ __forceinline__ void wait_async0() {
#if __has_builtin(__builtin_amdgcn_s_wait_asynccnt)
    __builtin_amdgcn_s_wait_asynccnt(0);
#else
    asm volatile("s_wait_asynccnt 0x0" ::: "memory");
#endif
}

__global__ __launch_bounds__(NTHREADS, 1)
void noisy_linear_wmma_f32(const float* __restrict__ X,
                           const float* __restrict__ W,
                           const float* __restrict__ Bias,
                           const float* __restrict__ Wn,
                           const float* __restrict__ Bn,
                           float* __restrict__ Out)
{
    __shared__ float sA[2][BM * LDT];   // 2 * 128*20*4 = 20 KB
    __shared__ float sB[2][BN * LDT];   // 2 * 256*20*4 = 40 KB

    const int tid  = threadIdx.x;
    const int lane = tid & 31;
    const int wid  = tid >> 5;     // 0..7
    const int wm   = wid & 1;      // wave M offset: 64*wm
    const int wn   = wid >> 1;     // wave N offset: 64*wn
    const int r    = lane & 15;    // row within 16x16 fragment
    const int hi   = lane >> 4;    // K-pair selector for A/B operands

    const int m0 = blockIdx.y * BM;
    const int n0 = blockIdx.x * BN;

    // B staging registers: 256 rows * 16 floats = 1024 float4 -> 4 per thread
    v4f rb[4];

    v8f acc[4][4];
#pragma unroll
    for (int i = 0; i < 4; ++i)
#pragma unroll
        for (int j = 0; j < 4; ++j)
            acc[i][j] = (v8f)0.0f;

    const int KT = K_DIM / BK;     // 256 K-tiles

    // ---------- prologue: k-tile 0 ----------
    // A tile: 128 rows * 16 floats = 512 float4 -> 2 async b128 per thread
#pragma unroll
    for (int i = 0; i < 2; ++i) {
        int idx = tid + i * NTHREADS;          // 0..511
        int row = idx >> 2;                    // 4 float4 per row
        int c4  = idx & 3;
        async_copy_b128(X + (size_t)(m0 + row) * K_DIM + c4 * 4,
                        &sA[0][row * LDT + c4 * 4]);
    }
#pragma unroll
    for (int i = 0; i < 4; ++i) {
        int idx = tid + i * NTHREADS;          // 0..1023
        int row = idx >> 2;
        int c4  = idx & 3;
        size_t g = (size_t)(n0 + row) * K_DIM + c4 * 4;
        v4f w  = *(const v4f*)(W  + g);
        v4f nw = *(const v4f*)(Wn + g);
        rb[i] = w + nw;                        // fuse noisy-weight add
    }
#pragma unroll
    for (int i = 0; i < 4; ++i) {
        int idx = tid + i * NTHREADS;
        int row = idx >> 2;
        int c4  = idx & 3;
        *(v4f*)&sB[0][row * LDT + c4 * 4] = rb[i];
    }
    wait_async0();
    __syncthreads();

    // ---------- main loop: software-pipelined double buffer ----------
    for (int kb = 0; kb < KT; ++kb) {
        const int nxt = kb + 1;
        const int cur = kb & 1;
        const int nb  = nxt & 1;

        if (nxt < KT) {
            const int koff = nxt * BK;
            // issue async A loads for the next buffer first (max overlap)
#pragma unroll
            for (int i = 0; i < 2; ++i) {
                int idx = tid + i * NTHREADS;
                int row = idx >> 2;
                int c4  = idx & 3;
                async_copy_b128(X + (size_t)(m0 + row) * K_DIM + koff + c4 * 4,
                                &sA[nb][row * LDT + c4 * 4]);
            }
#pragma unroll
            for (int i = 0; i < 4; ++i) {
                int idx = tid + i * NTHREADS;
                int row = idx >> 2;
                int c4  = idx & 3;
                size_t g = (size_t)(n0 + row) * K_DIM + koff + c4 * 4;
                v4f w  = *(const v4f*)(W  + g);
                v4f nw = *(const v4f*)(Wn + g);
                rb[i] = w + nw;
            }
        }

        // prefetch the big weight streams two K-tiles ahead (one row / thread)
        if (kb + 2 < KT) {
            const int kpf = (kb + 2) * BK;
            size_t gb = (size_t)(n0 + tid) * K_DIM + kpf;   // B rows: 256 threads cover tile
            __builtin_prefetch(W  + gb, 0, 1);
            __builtin_prefetch(Wn + gb, 0, 1);
            if (tid < BM) {
                size_t ga = (size_t)(m0 + tid) * K_DIM + kpf;
                __builtin_prefetch(X + ga, 0, 1);
            }
        }

        // ---- compute on buffer `cur`: 4 K-steps of 16 WMMAs ----
        {
            const float* pa = &sA[cur][0];
            const float* pb = &sB[cur][0];
#pragma unroll
            for (int kk = 0; kk < BK; kk += 4) {
                v2f af[4], bf[4];
#pragma unroll
                for (int i = 0; i < 4; ++i)
                    af[i] = *(const v2f*)(pa + (wm * 64 + i * 16 + r) * LDT + kk + hi * 2);
#pragma unroll
                for (int j = 0; j < 4; ++j)
                    bf[j] = *(const v2f*)(pb + (wn * 64 + j * 16 + r) * LDT + kk + hi * 2);
#pragma unroll
                for (int i = 0; i < 4; ++i)
#pragma unroll
                    for (int j = 0; j < 4; ++j)
                        acc[i][j] = __builtin_amdgcn_wmma_f32_16x16x4_f32(
                            /*neg_a=*/false, af[i],
                            /*neg_b=*/false, bf[j],
                            /*c_mod=*/(short)0, acc[i][j],
                            /*reuse_a=*/false, /*reuse_b=*/false);
            }
        }

        if (nxt < KT) {
#pragma unroll
            for (int i = 0; i < 4; ++i) {
                int idx = tid + i * NTHREADS;
                int row = idx >> 2;
                int c4  = idx & 3;
                *(v4f*)&sB[nb][row * LDT + c4 * 4] = rb[i];
            }
        }
        // my async A loads for `nb` must land before anyone computes from it
        wait_async0();
        __syncthreads();
    }

    // ---------- epilogue: fused bias + bias_noise ----------
#pragma unroll
    for (int j = 0; j < 4; ++j) {
        const int col = n0 + wn * 64 + j * 16 + r;
        const float badd = Bias[col] + Bn[col];
#pragma unroll
        for (int i = 0; i < 4; ++i) {
            const int rowbase = m0 + wm * 64 + i * 16 + hi * 8;
#pragma unroll
            for (int rr = 0; rr < 8; ++rr) {
                Out[(size_t)(rowbase + rr) * N_DIM + col] = acc[i][j][rr] + badd;
            }
        }
    }
}

extern "C" void kernel_launch(void* const* d_in, const int* in_sizes, int n_in,
                              void* d_out, int out_size, void* d_ws, size_t ws_size,
                              hipStream_t stream) {
    const float* x  = (const float*)d_in[0];   // [2048, 4096]
    const float* w  = (const float*)d_in[1];   // [4096, 4096]
    const float* b  = (const float*)d_in[2];   // [4096]
    const float* wn = (const float*)d_in[3];   // [4096, 4096]
    const float* bn = (const float*)d_in[4];   // [4096]
    float* out = (float*)d_out;                // [2048, 4096]

    dim3 grid(N_DIM / BN, M_DIM / BM);         // (16, 16)
    dim3 block(NTHREADS);                      // 256 threads = 8 wave32
    noisy_linear_wmma_f32<<<grid, block, 0, stream>>>(x, w, b, wn, bn, out);
}